// MGNO_EncoderDecoder_Block_39152921871007
// MI455X (gfx1250) — compile-verified
//
#include <hip/hip_runtime.h>
#include <hip/hip_bf16.h>

// ---------------------------------------------------------------------------
// MGNO encoder/decoder block for MI455X (gfx1250, wave32, WMMA + TDM).
// Heavy GEMMs: v_wmma_f32_16x16x4_f32 with tensor_load_to_lds (TDM) staging,
// double-buffered in LDS, DMA overlapped with WMMA compute.
// ---------------------------------------------------------------------------

typedef __attribute__((ext_vector_type(2))) float v2f;
typedef __attribute__((ext_vector_type(8))) float v8f;
typedef __attribute__((ext_vector_type(4))) unsigned int u32x4;
typedef __attribute__((ext_vector_type(4))) int i32x4;
typedef __attribute__((ext_vector_type(8))) int i32x8;

#define TILE_M 64
#define TILE_N 64
#define TILE_K 32

// Problem constants
#define LVL 4
#define NTOK 16384            // b*n*nv
#define NTOK4 65536           // L * NTOK
#define CATT 1024
#define COUT 256
#define CLVL 256
#define ATTD 128
#define LVL_STRIDE 4194304    // b*n*nv*c = stride between levels in x_levels

// ---------------------------------------------------------------------------
// TDM: issue a 2-D tensor_load_to_lds of a tileX x tileY (elements x rows)
// fp32 tile, global row stride strideElts, into LDS at byte offset ldsOff.
// padCode: pad_interval code (4 -> every 32 DWORDs, 5 -> every 64 DWORDs),
// pad_amount = 1 DWORD -> LDS row strides of 33 / 65 floats (bank-conflict-free).
// ---------------------------------------------------------------------------
__device__ __forceinline__ void tdm_load_2d(unsigned int ldsOff,
                                            const float* g,
                                            int tileX, int tileY,
                                            unsigned long long strideElts,
                                            unsigned int padCode) {
  const unsigned long long ga = (unsigned long long)(uintptr_t)g;
  u32x4 g0;
  g0[0] = 1u;                                      // count=1 valid descriptor
  g0[1] = ldsOff;                                  // lds_addr (bytes)
  g0[2] = (unsigned int)ga;                        // global_addr[31:0]
  g0[3] = (unsigned int)((ga >> 32) & 0x1FFFFFFull) | (2u << 30); // [56:32] | type=2

  const unsigned int td0 = (unsigned int)tileX;    // tensor_dim0 (no OOB clip)
  const unsigned int td1 = (unsigned int)tileY;    // tensor_dim1
  i32x8 g1;
  g1[0] = (int)((2u << 16) | (1u << 20) | (padCode << 22)); // data_size=4B, pad_enable, pad_interval, pad_amount=0(+1 dw)
  g1[1] = (int)((td0 & 0xFFFFu) << 16);                     // tensor_dim0[15:0] @ bits 63:48
  g1[2] = (int)((td0 >> 16) | ((td1 & 0xFFFFu) << 16));     // td0[31:16] | td1[15:0]
  g1[3] = (int)((td1 >> 16) | ((unsigned int)tileX << 16)); // td1[31:16] | tile_dim0
  g1[4] = (int)((unsigned int)tileY & 0xFFFFu);             // tile_dim1 (tile_dim2=0)
  g1[5] = (int)(unsigned int)(strideElts & 0xFFFFFFFFull);  // tensor_dim0_stride[31:0]
  g1[6] = (int)(unsigned int)((strideElts >> 32) & 0xFFFFull); // stride[47:32] | dim1_stride=0
  g1[7] = 0;

  const i32x4 gz = {0, 0, 0, 0};                   // groups 2/3: zero (2-D tensor)
#if defined(__clang_major__) && (__clang_major__ >= 23)
  const i32x8 gz8 = {0, 0, 0, 0, 0, 0, 0, 0};
  __builtin_amdgcn_tensor_load_to_lds(g0, g1, gz, gz, gz8, 0);
#else
  __builtin_amdgcn_tensor_load_to_lds(g0, g1, gz, gz, 0);
#endif
}

struct GemmP {
  const float* A; const float* B; float* C;
  int M, N, K;
  int aRow, aChunk;              // A(r,kk) = A + (kk>>8)*aChunk + r*aRow + (kk&255)
  int bLevelStride, rowsPerLevel;// B base += (rowBase/rowsPerLevel)*bLevelStride
  const float* bias;             // EPI 0 (optional), 1, 3
  const float* skip;             // EPI 2, 3
  const float* scale;            // EPI 2 (gamma, idx (row&3)*256+col), 3 (gamma_mlp[col])
};

// EPI: 0 = acc (+bias)            (plain store)
//      1 = silu(acc + bias)       (MLP hidden)
//      2 = skip + gamma*acc       (attention residual merge)
//      3 = skip + scale*(acc+bias)(final output)
template <int EPI>
__global__ __launch_bounds__(128) void gemm_wmma(GemmP p) {
  __shared__ float As[2][TILE_M][TILE_K + 1];
  __shared__ float Bs[2][TILE_K][TILE_N + 1];

  const int tid  = threadIdx.x;
  const int lane = tid & 31;
  const int wv   = tid >> 5;       // 4 waves
  const int wm   = wv >> 1;        // 0..1 : 32-row half
  const int wn   = wv & 1;         // 0..1 : 32-col half
  const int tm   = lane & 15;
  const int khi  = (lane >> 4) << 1;   // 0 or 2 (K sub-offset per 16x16x4 layout)

  const int rowBase = blockIdx.y * TILE_M;
  const int colBase = blockIdx.x * TILE_N;
  const float* Bp = p.B + (size_t)(rowBase / p.rowsPerLevel) * p.bLevelStride;

  const int T = p.K / TILE_K;

  // TDM issue for tile t into LDS buffer buf (wave 0 only, one DMA per tile).
  auto issue = [&](int t, int buf) {
    const int kt = t * TILE_K;
    const float* aG = p.A + (size_t)(kt >> 8) * p.aChunk +
                      (size_t)rowBase * p.aRow + (kt & 255);
    const float* bG = Bp + (size_t)kt * p.N + colBase;
    tdm_load_2d((unsigned int)(uintptr_t)&As[buf][0][0], aG,
                TILE_K, TILE_M, (unsigned long long)p.aRow, 4u);  // 32-dw rows -> stride 33
    tdm_load_2d((unsigned int)(uintptr_t)&Bs[buf][0][0], bG,
                TILE_N, TILE_K, (unsigned long long)p.N, 5u);     // 64-dw rows -> stride 65
  };

  v8f acc[2][2] = {};

  if (wv == 0) issue(0, 0);

  for (int t = 0; t < T; ++t) {
    const int buf = t & 1;
    if (wv == 0) __builtin_amdgcn_s_wait_tensorcnt(0);  // tile t landed in LDS
    __syncthreads();                                    // visible to all waves
    if (wv == 0 && (t + 1) < T) issue(t + 1, buf ^ 1);  // DMA next tile over compute

#pragma unroll
    for (int kk = 0; kk < TILE_K; kk += 4) {
      v2f a0, a1, b0, b1;
      a0[0] = As[buf][(wm << 5) + tm][kk + khi];
      a0[1] = As[buf][(wm << 5) + tm][kk + khi + 1];
      a1[0] = As[buf][(wm << 5) + 16 + tm][kk + khi];
      a1[1] = As[buf][(wm << 5) + 16 + tm][kk + khi + 1];
      b0[0] = Bs[buf][kk + khi][(wn << 5) + tm];
      b0[1] = Bs[buf][kk + khi + 1][(wn << 5) + tm];
      b1[0] = Bs[buf][kk + khi][(wn << 5) + 16 + tm];
      b1[1] = Bs[buf][kk + khi + 1][(wn << 5) + 16 + tm];
      acc[0][0] = __builtin_amdgcn_wmma_f32_16x16x4_f32(false, a0, false, b0, (short)0, acc[0][0], false, false);
      acc[0][1] = __builtin_amdgcn_wmma_f32_16x16x4_f32(false, a0, false, b1, (short)0, acc[0][1], false, false);
      acc[1][0] = __builtin_amdgcn_wmma_f32_16x16x4_f32(false, a1, false, b0, (short)0, acc[1][0], false, false);
      acc[1][1] = __builtin_amdgcn_wmma_f32_16x16x4_f32(false, a1, false, b1, (short)0, acc[1][1], false, false);
    }
    __syncthreads();                                    // compute done; buf reusable
  }

  // C/D layout: VGPR e -> M = e + (lane>=16 ? 8 : 0), N = lane&15
  const int rsub = (lane >> 4) << 3;
#pragma unroll
  for (int i = 0; i < 2; ++i) {
#pragma unroll
    for (int j = 0; j < 2; ++j) {
#pragma unroll
      for (int e = 0; e < 8; ++e) {
        const int row = rowBase + (wm << 5) + (i << 4) + rsub + e;
        const int col = colBase + (wn << 5) + (j << 4) + tm;
        const size_t idx = (size_t)row * p.N + col;
        float a = acc[i][j][e];
        if (EPI == 0) {
          if (p.bias) a += p.bias[col];
          p.C[idx] = a;
        } else if (EPI == 1) {
          a += p.bias[col];
          p.C[idx] = a / (1.0f + __expf(-a));
        } else if (EPI == 2) {
          p.C[idx] = p.skip[idx] + p.scale[((row & 3) << 8) + col] * a;
        } else {
          a += p.bias[col];
          p.C[idx] = p.skip[idx] + p.scale[col] * a;
        }
      }
    }
  }
}

// ------------------------- adaLN modulation (tiny) -------------------------
__global__ __launch_bounds__(256) void mod_kernel(const float* __restrict__ gt,
                                                  const int* __restrict__ gidx,
                                                  const float* __restrict__ W_ada,
                                                  const float* __restrict__ b_ada,
                                                  float* __restrict__ mod) {
  const int l = blockIdx.x;
  const int tid = threadIdx.x;
  __shared__ float g[128];
  if (tid < 128) g[tid] = gt[gidx[l] * 128 + tid];
  __syncthreads();
#pragma unroll
  for (int u = 0; u < 8; ++u) {
    const int d = tid + (u << 8);
    float acc = b_ada[l * 2048 + d];
    for (int e = 0; e < 128; ++e)
      acc += g[e] * W_ada[((size_t)l * 128 + e) * 2048 + d];
    mod[l * 2048 + d] = acc;
  }
}

// ----------------------------- LayerNorm ----------------------------------
// ADA: y = nrm*(1+mod[l][ch]) + mod[l][1024+ch], l = row>>14
// else: y = nrm*P0[ch] + P1[ch]
template <bool ADA>
__global__ __launch_bounds__(256) void ln_kernel(const float* __restrict__ X,
                                                 float* __restrict__ Y,
                                                 const float* __restrict__ P0,
                                                 const float* __restrict__ P1) {
  const int row = blockIdx.x;
  const int tid = threadIdx.x;
  const float* x = X + (size_t)row * CATT;
  float v[4];
  float s = 0.0f, ss = 0.0f;
#pragma unroll
  for (int i = 0; i < 4; ++i) {
    v[i] = x[tid + (i << 8)];
    s += v[i];
    ss += v[i] * v[i];
  }
#pragma unroll
  for (int off = 16; off >= 1; off >>= 1) {
    s  += __shfl_down(s, off, 32);
    ss += __shfl_down(ss, off, 32);
  }
  __shared__ float sh[16];
  if ((tid & 31) == 0) { sh[tid >> 5] = s; sh[8 + (tid >> 5)] = ss; }
  __syncthreads();
  s = 0.0f; ss = 0.0f;
#pragma unroll
  for (int w = 0; w < 8; ++w) { s += sh[w]; ss += sh[8 + w]; }
  const float mean = s * (1.0f / 1024.0f);
  const float var  = ss * (1.0f / 1024.0f) - mean * mean;
  const float rstd = rsqrtf(var + 1e-5f);
#pragma unroll
  for (int i = 0; i < 4; ++i) {
    const int ch = tid + (i << 8);
    const float nrm = (v[i] - mean) * rstd;
    float y;
    if (ADA) {
      const int l = row >> 14;
      y = nrm * (1.0f + P0[l * 2048 + ch]) + P0[l * 2048 + 1024 + ch];
    } else {
      y = nrm * P0[ch] + P1[ch];
    }
    Y[(size_t)row * CATT + ch] = y;
  }
}

// --------------------- attention over the L=4 grid axis --------------------
// q,k,v: [L*NTOK, 128] with row = l*NTOK + r; out o: [NTOK*4, 128], row = r*4+l
__global__ __launch_bounds__(256) void attn_kernel(const float* __restrict__ q,
                                                   const float* __restrict__ kx,
                                                   const float* __restrict__ vx,
                                                   float* __restrict__ o) {
  const int gid = blockIdx.x * 256 + threadIdx.x;  // NTOK*8 threads
  const int r = gid >> 3;
  const int h = gid & 7;
  float kb[4][16], vb[4][16];
#pragma unroll
  for (int l = 0; l < 4; ++l) {
    const size_t base = ((size_t)((l << 14) + r)) * ATTD + (h << 4);
    const float4* kp = reinterpret_cast<const float4*>(kx + base);
    const float4* vp = reinterpret_cast<const float4*>(vx + base);
#pragma unroll
    for (int t = 0; t < 4; ++t) {
      const float4 kk4 = kp[t]; const float4 vv4 = vp[t];
      kb[l][t * 4 + 0] = kk4.x; kb[l][t * 4 + 1] = kk4.y;
      kb[l][t * 4 + 2] = kk4.z; kb[l][t * 4 + 3] = kk4.w;
      vb[l][t * 4 + 0] = vv4.x; vb[l][t * 4 + 1] = vv4.y;
      vb[l][t * 4 + 2] = vv4.z; vb[l][t * 4 + 3] = vv4.w;
    }
  }
#pragma unroll
  for (int i = 0; i < 4; ++i) {
    float qb[16];
    const size_t qbase = ((size_t)((i << 14) + r)) * ATTD + (h << 4);
    const float4* qp = reinterpret_cast<const float4*>(q + qbase);
#pragma unroll
    for (int t = 0; t < 4; ++t) {
      const float4 q4 = qp[t];
      qb[t * 4 + 0] = q4.x; qb[t * 4 + 1] = q4.y;
      qb[t * 4 + 2] = q4.z; qb[t * 4 + 3] = q4.w;
    }
    float sc[4];
#pragma unroll
    for (int j = 0; j < 4; ++j) {
      float d = 0.0f;
#pragma unroll
      for (int t = 0; t < 16; ++t) d += qb[t] * kb[j][t];
      sc[j] = d * 0.25f;  // 1/sqrt(16)
    }
    float m = sc[0];
#pragma unroll
    for (int j = 1; j < 4; ++j) m = fmaxf(m, sc[j]);
    float e[4], sum = 0.0f;
#pragma unroll
    for (int j = 0; j < 4; ++j) { e[j] = __expf(sc[j] - m); sum += e[j]; }
    const float inv = 1.0f / sum;
    float ob[16];
#pragma unroll
    for (int t = 0; t < 16; ++t) {
      float a = 0.0f;
#pragma unroll
      for (int j = 0; j < 4; ++j) a += e[j] * vb[j][t];
      ob[t] = a * inv;
    }
    float* op = o + ((size_t)((r << 2) + i)) * ATTD + (h << 4);
    float4* op4 = reinterpret_cast<float4*>(op);
#pragma unroll
    for (int t = 0; t < 4; ++t)
      op4[t] = make_float4(ob[t * 4], ob[t * 4 + 1], ob[t * 4 + 2], ob[t * 4 + 3]);
  }
}

// ---------------------------------------------------------------------------
static inline void launch_gemm(int epi, const GemmP& p, hipStream_t s) {
  dim3 g(p.N / TILE_N, p.M / TILE_M), b(128);
  switch (epi) {
    case 0: gemm_wmma<0><<<g, b, 0, s>>>(p); break;
    case 1: gemm_wmma<1><<<g, b, 0, s>>>(p); break;
    case 2: gemm_wmma<2><<<g, b, 0, s>>>(p); break;
    default: gemm_wmma<3><<<g, b, 0, s>>>(p); break;
  }
}

extern "C" void kernel_launch(void* const* d_in, const int* in_sizes, int n_in,
                              void* d_out, int out_size, void* d_ws, size_t ws_size,
                              hipStream_t stream) {
  (void)in_sizes; (void)n_in; (void)out_size; (void)ws_size;
  const float* x_levels   = (const float*)d_in[0];
  const float* W_skip     = (const float*)d_in[1];
  const float* b_skip     = (const float*)d_in[2];
  const float* Wp         = (const float*)d_in[3];
  const float* grid_table = (const float*)d_in[4];
  const float* W_ada      = (const float*)d_in[5];
  const float* b_ada      = (const float*)d_in[6];
  const float* Wq         = (const float*)d_in[7];
  const float* Wk         = (const float*)d_in[8];
  const float* Wv         = (const float*)d_in[9];
  const float* Wo         = (const float*)d_in[10];
  const float* gamma      = (const float*)d_in[11];
  const float* W_skip_mlp = (const float*)d_in[12];
  const float* b_skip_mlp = (const float*)d_in[13];
  const float* ln_w       = (const float*)d_in[14];
  const float* ln_b       = (const float*)d_in[15];
  const float* W1         = (const float*)d_in[16];
  const float* b1         = (const float*)d_in[17];
  const float* W2         = (const float*)d_in[18];
  const float* b2         = (const float*)d_in[19];
  const float* gamma_mlp  = (const float*)d_in[20];
  const int*   grid_idx   = (const int*)d_in[21];
  float* out = (float*)d_out;
  float* ws  = (float*)d_ws;

  // workspace layout (floats), with deliberate aliasing of dead buffers
  const size_t oXP    = 0;                              // [65536,1024] xp (later: xres)
  const size_t oXN    = oXP + (size_t)NTOK4 * CATT;     // [65536,1024] xn (later: o_att @0, xn2 @+8M)
  const size_t oSKIP  = oXN + (size_t)NTOK4 * CATT;     // [16384,1024] x_skip (later: h)
  const size_t oQ     = oSKIP + (size_t)NTOK * CATT;    // [65536,128]
  const size_t oK     = oQ + (size_t)NTOK4 * ATTD;
  const size_t oV     = oK + (size_t)NTOK4 * ATTD;
  const size_t oSKIP2 = oV + (size_t)NTOK4 * ATTD;      // [16384,256]
  const size_t oMOD   = oSKIP2 + (size_t)NTOK * COUT;   // [4,2048]
  const size_t oO     = oXN;                            // [65536,128] o_att
  const size_t oXN2   = oXN + (size_t)NTOK4 * ATTD;     // [16384,1024]
  const size_t oXRES  = oXP;                            // [16384,1024]
  const size_t oH     = oSKIP;                          // [16384,1024]

  const int BIG = 1 << 30;

  // 0) adaLN modulation (tiny)
  mod_kernel<<<dim3(LVL), dim3(256), 0, stream>>>(grid_table, grid_idx, W_ada, b_ada, ws + oMOD);

  // 1) x_skip = x_cat @ W_skip + b_skip   (level-chunked A view of x_levels)
  { GemmP p{x_levels, W_skip, ws + oSKIP, NTOK, CATT, CATT,
            CLVL, LVL_STRIDE, 0, BIG, b_skip, nullptr, nullptr};
    launch_gemm(0, p, stream); }

  // 2) xp[l] = x_levels[l] @ Wp[l]   (one GEMM, B selected per 16384-row level)
  { GemmP p{x_levels, Wp, ws + oXP, NTOK4, CATT, CLVL,
            CLVL, CLVL, CLVL * CATT, NTOK, nullptr, nullptr, nullptr};
    launch_gemm(0, p, stream); }

  // 3) adaptive LN: xn = LN(xp)*(1+scale[l]) + shift[l]
  ln_kernel<true><<<dim3(NTOK4), dim3(256), 0, stream>>>(ws + oXP, ws + oXN, ws + oMOD, nullptr);

  // 4) q = xn @ Wq ; k = xn @ Wk ; v = xp @ Wv
  { GemmP p{ws + oXN, Wq, ws + oQ, NTOK4, ATTD, CATT,
            CATT, CLVL, 0, BIG, nullptr, nullptr, nullptr};
    launch_gemm(0, p, stream); }
  { GemmP p{ws + oXN, Wk, ws + oK, NTOK4, ATTD, CATT,
            CATT, CLVL, 0, BIG, nullptr, nullptr, nullptr};
    launch_gemm(0, p, stream); }
  { GemmP p{ws + oXP, Wv, ws + oV, NTOK4, ATTD, CATT,
            CATT, CLVL, 0, BIG, nullptr, nullptr, nullptr};
    launch_gemm(0, p, stream); }

  // 5) attention over the grid/level axis (L=4 tokens, 8 heads, dh=16)
  attn_kernel<<<dim3((NTOK * 8) / 256), dim3(256), 0, stream>>>(
      ws + oQ, ws + oK, ws + oV, ws + oO);

  // 6) x = x_skip + gamma * (o @ Wo)    -> xres [16384,1024] (as [65536,256])
  { GemmP p{ws + oO, Wo, ws + oXRES, NTOK4, COUT, ATTD,
            ATTD, CLVL, 0, BIG, nullptr, ws + oSKIP, gamma};
    launch_gemm(2, p, stream); }

  // 7) x_skip2 = x @ W_skip_mlp + b
  { GemmP p{ws + oXRES, W_skip_mlp, ws + oSKIP2, NTOK, COUT, CATT,
            CATT, CLVL, 0, BIG, b_skip_mlp, nullptr, nullptr};
    launch_gemm(0, p, stream); }

  // 8) xn2 = LN(x)*ln_w + ln_b
  ln_kernel<false><<<dim3(NTOK), dim3(256), 0, stream>>>(ws + oXRES, ws + oXN2, ln_w, ln_b);

  // 9) h = silu(xn2 @ W1 + b1)
  { GemmP p{ws + oXN2, W1, ws + oH, NTOK, CATT, CATT,
            CATT, CLVL, 0, BIG, b1, nullptr, nullptr};
    launch_gemm(1, p, stream); }

  // 10) out = x_skip2 + gamma_mlp * (h @ W2 + b2)
  { GemmP p{ws + oH, W2, out, NTOK, COUT, CATT,
            CATT, CLVL, 0, BIG, b2, ws + oSKIP2, gamma_mlp};
    launch_gemm(3, p, stream); }
}